// GPR_sparse_32126355374954
// MI455X (gfx1250) — compile-verified
//
#include <hip/hip_runtime.h>
#include <hip/hip_bf16.h>

typedef __attribute__((ext_vector_type(2))) float v2f;
typedef __attribute__((ext_vector_type(8))) float v8f;

#define ND 64   // feature dim

// ---------------------------------------------------------------------------
// init: hidden = x*temp[0]; x_cur = x; x_new = 0
// ---------------------------------------------------------------------------
__global__ __launch_bounds__(256) void gpr_init_kernel(
    const float* __restrict__ x, const float* __restrict__ temp,
    float* __restrict__ hidden, float* __restrict__ xcur,
    float* __restrict__ xnew, int total) {
  int i = blockIdx.x * 256 + threadIdx.x;
  if (i < total) {
    float v = x[i];
    hidden[i] = v * temp[0];
    xcur[i]   = v;
    xnew[i]   = 0.0f;
  }
}

// ---------------------------------------------------------------------------
// GEMM: H = Xcur @ W_i^T + b_i  (N x 64) using V_WMMA_F32_16X16X4_F32.
// One wave32 per 16x16 output tile; K=64 walked in 16 steps of 4.
// A (16x4 f32): lanes 0-15 hold M=lane K{0,1}; lanes 16-31 hold M=lane-16 K{2,3}
// B (4x16 f32): lanes 0-15 hold K{0,1} col=lane; lanes 16-31 hold K{2,3}
// -> both are contiguous float2 loads from row-major X / W.
// C/D (16x16 f32): VGPR v, lane-half h -> row M = v + 8*h, col N = lane&15
// ---------------------------------------------------------------------------
__global__ __launch_bounds__(256) void gpr_gemm_wmma_kernel(
    const float* __restrict__ X, const float* __restrict__ Wm,
    const float* __restrict__ bias, float* __restrict__ H,
    int nWaves) {
  int wave = (blockIdx.x * 256 + threadIdx.x) >> 5;
  if (wave >= nWaves) return;            // wave-uniform exit, EXEC stays all-1s
  int lane  = threadIdx.x & 31;
  int mtile = wave >> 2;                 // N/16 row tiles
  int otile = wave & 3;                  // 64/16 col tiles
  int hi = lane >> 4;                    // lane half
  int lo = lane & 15;

  const float* arow = X  + (size_t)(mtile * 16 + lo) * ND;  // A row M=lo
  const float* brow = Wm + (size_t)(otile * 16 + lo) * ND;  // B col o=lo (W[o][d])

  v8f c = {};
#pragma unroll
  for (int k = 0; k < ND; k += 4) {
    v2f a = *(const v2f*)(arow + k + 2 * hi);
    v2f b = *(const v2f*)(brow + k + 2 * hi);
    // D = A x B + C, native fp32 WMMA
    c = __builtin_amdgcn_wmma_f32_16x16x4_f32(
        /*neg_a=*/false, a, /*neg_b=*/false, b,
        /*c_mod=*/(short)0, c, /*reuse_a=*/false, /*reuse_b=*/false);
  }

  float bv = bias[otile * 16 + lo];
  float* hout = H + (size_t)(mtile * 16 + 8 * hi) * ND + otile * 16 + lo;
#pragma unroll
  for (int v = 0; v < 8; ++v)
    hout[(size_t)v * ND] = c[v] + bv;
}

// ---------------------------------------------------------------------------
// Scatter: for each edge e: xnew[dst[e]] += h[src[e]] * w[e]
// 16 lanes per edge, one float4 (b128) gather per lane half-row,
// 4 hardware f32 atomics per lane into the L2-resident accumulator.
// ---------------------------------------------------------------------------
__global__ __launch_bounds__(256) void gpr_scatter_kernel(
    const float* __restrict__ H, const float* __restrict__ w,
    const int* __restrict__ src, const int* __restrict__ dst,
    float* __restrict__ xnew, int nEdges) {
  int tid = blockIdx.x * 256 + threadIdx.x;
  int e = tid >> 4;
  if (e >= nEdges) return;
  int sub = tid & 15;

  int   s  = src[e];
  int   d  = dst[e];
  float we = w[e];

  const float4* hp = (const float4*)(H + (size_t)s * ND) + sub;
  float4 hv = *hp;

  float* out = xnew + (size_t)d * ND + sub * 4;
  unsafeAtomicAdd(out + 0, hv.x * we);
  unsafeAtomicAdd(out + 1, hv.y * we);
  unsafeAtomicAdd(out + 2, hv.z * we);
  unsafeAtomicAdd(out + 3, hv.w * we);
}

// ---------------------------------------------------------------------------
// Finalize layer i: v = relu(xnew); xcur = v; hidden += v*temp[i+1]; xnew = 0
// (re-zeroing here feeds the next layer's scatter; init re-zeroes at call start
//  so replayed graphs see identical state)
// ---------------------------------------------------------------------------
__global__ __launch_bounds__(256) void gpr_finalize_kernel(
    float* __restrict__ xnew, float* __restrict__ xcur,
    float* __restrict__ hidden, const float* __restrict__ temp,
    int tempIdx, int total) {
  int i = blockIdx.x * 256 + threadIdx.x;
  if (i < total) {
    float v = fmaxf(xnew[i], 0.0f);
    xcur[i] = v;
    hidden[i] += v * temp[tempIdx];
    xnew[i] = 0.0f;
  }
}

// ---------------------------------------------------------------------------
extern "C" void kernel_launch(void* const* d_in, const int* in_sizes, int n_in,
                              void* d_out, int out_size, void* d_ws, size_t ws_size,
                              hipStream_t stream) {
  const float* x    = (const float*)d_in[0];  // [N, 64]
  const float* w    = (const float*)d_in[1];  // [E]
  const float* Wm   = (const float*)d_in[2];  // [L, 64, 64]
  const float* bias = (const float*)d_in[3];  // [L, 64]
  const float* temp = (const float*)d_in[4];  // [L+1]
  const int*   src  = (const int*)d_in[5];    // [E]
  const int*   dst  = (const int*)d_in[6];    // [E]
  float* hidden = (float*)d_out;              // [N, 64]

  const int n = in_sizes[0] / ND;             // 50000
  const int E = in_sizes[1];                  // 1600000
  const int L = in_sizes[3] / ND;             // 4
  const int total = n * ND;

  float* h    = (float*)d_ws;                 // [N,64]
  float* xcur = h    + (size_t)total;         // [N,64]
  float* xnew = xcur + (size_t)total;         // [N,64]

  const int ewBlocks   = (total + 255) / 256;
  const int nWaves     = (n / 16) * 4;                      // 16-row x 16-col tiles
  const int gemmBlocks = (nWaves * 32 + 255) / 256;
  const int scatBlocks = (E * 16 + 255) / 256;

  gpr_init_kernel<<<ewBlocks, 256, 0, stream>>>(x, temp, hidden, xcur, xnew, total);

  for (int i = 0; i < L; ++i) {
    gpr_gemm_wmma_kernel<<<gemmBlocks, 256, 0, stream>>>(
        xcur, Wm + (size_t)i * ND * ND, bias + (size_t)i * ND, h, nWaves);
    gpr_scatter_kernel<<<scatBlocks, 256, 0, stream>>>(h, w, src, dst, xnew, E);
    gpr_finalize_kernel<<<ewBlocks, 256, 0, stream>>>(
        xnew, xcur, hidden, temp, i + 1, total);
  }
}